// Temporal_67482526155423
// MI455X (gfx1250) — compile-verified
//
#include <hip/hip_runtime.h>
#include <stdint.h>

// ---------------------------------------------------------------------------
// Spike encoder: out[t, i] = (t >= 64*x[i]) * Bernoulli(0.75)
//   x:   N = 16*3*224*224 = 2,408,448 fp32   (~9.6 MB)
//   out: 64 * N fp32                          (~616.6 MB)  -> store-bandwidth bound
//
// MI455X plan:
//  * memory floor: ~626 MB / 23.3 TB/s ~= 27 us. Everything else must hide
//    under that, so: B128 NT stores for the output stream (never re-read,
//    bigger than the 192MB L2), and a cheap exact-Bernoulli RNG.
//  * RNG: p=0.75 is exact with 2 bits (P(2 bits != 0) = 3/4). One 20-round
//    threefry2x32 call => 64 bits => 32 draws. 8 calls cover a thread's
//    4 elements x 64 timesteps.
//  * CDNA5 paths: per-wave TDM tensor_load_to_lds stages the wave's 512B
//    input slice into LDS (TENSORcnt + s_wait_tensorcnt), threads read it
//    back with ds_load_b128. No barrier needed: each wave reads only the
//    LDS region it DMA'd itself.
//  * WMMA: zero matmul FLOPs in this op -> matrix pipes are not applicable.
// ---------------------------------------------------------------------------

typedef __attribute__((ext_vector_type(4))) float        f32x4;
typedef __attribute__((ext_vector_type(4))) unsigned int u32x4;
typedef __attribute__((ext_vector_type(8))) unsigned int u32x8;

#define TIME_STEPS 64

__device__ __forceinline__ uint32_t rotl32(uint32_t v, uint32_t n) {
  return (v << n) | (v >> (32u - n));
}

// Threefry-2x32, 20 rounds (JAX's PRNG core).
__device__ __forceinline__ void threefry2x32(uint32_t k0, uint32_t k1,
                                             uint32_t c0, uint32_t c1,
                                             uint32_t& o0, uint32_t& o1) {
  const uint32_t R[8] = {13u, 15u, 26u, 6u, 17u, 29u, 16u, 24u};
  const uint32_t ks[3] = {k0, k1, k0 ^ k1 ^ 0x1BD11BDAu};
  uint32_t x0 = c0 + ks[0];
  uint32_t x1 = c1 + ks[1];
#pragma unroll
  for (int chunk = 0; chunk < 5; ++chunk) {
#pragma unroll
    for (int r = 0; r < 4; ++r) {
      x0 += x1;
      x1 = rotl32(x1, R[(chunk & 1) * 4 + r]);
      x1 ^= x0;
    }
    x0 += ks[(chunk + 1) % 3];
    x1 += ks[(chunk + 2) % 3] + (uint32_t)(chunk + 1);
  }
  o0 = x0;
  o1 = x1;
}

__global__ __launch_bounds__(256) void spike_encode_kernel(
    const float* __restrict__ x, float* __restrict__ out, unsigned int n) {
  __shared__ float tile[1024];  // 4 KB: this block's slice of x (8 waves x 512B)

  const unsigned int tid  = threadIdx.x;
  const unsigned int wave =
      (unsigned int)__builtin_amdgcn_readfirstlane((int)(tid >> 5));
  const unsigned int waveElemBase = blockIdx.x * 1024u + wave * 128u;

  // ---- TDM: async-load this wave's 128-float slice into LDS --------------
  unsigned int remain = (waveElemBase < n) ? (n - waveElemBase) : 0u;
  if (remain > 128u) remain = 128u;
  if (remain != 0u) {
    // LDS byte offset of this wave's region (generic-ptr low 32 bits = LDS offset).
    const unsigned int ldsOff =
        (unsigned int)(uintptr_t)(&tile[0]) + wave * 512u;
    const unsigned long long ga =
        (unsigned long long)(uintptr_t)(x + waveElemBase);

    // D# group 0: count=1 | lds_addr | global_addr[56:0] | type=2
    u32x4 g0;
    g0.x = 0x1u;
    g0.y = ldsOff;
    g0.z = (unsigned int)(ga & 0xFFFFFFFFull);
    g0.w = ((unsigned int)(ga >> 32) & 0x01FFFFFFu) | 0x80000000u;

    // D# group 1: data_size=4B, 1-D tile: tile_dim0=128, tensor_dim0=remain
    // (OOB reads in a partial tail wave return zero), tensor_dim1=tile_dim1=1.
    u32x8 g1;
    g1.s0 = 0x00020000u;                               // data_size = 2 (4B)
    g1.s1 = (remain & 0xFFFFu) << 16;                  // tensor_dim0[15:0]
    g1.s2 = ((remain >> 16) & 0xFFFFu) | (1u << 16);   // dim0 hi | tensor_dim1=1
    g1.s3 = 128u << 16;                                // tile_dim0 = 128
    g1.s4 = 1u;                                        // tile_dim1 = 1
    g1.s5 = 128u;                                      // tensor_dim0_stride lo
    g1.s6 = 0u;
    g1.s7 = 0u;

    asm volatile("tensor_load_to_lds %0, %1" ::"s"(g0), "s"(g1) : "memory");
    __builtin_amdgcn_s_wait_tensorcnt(0);
  }

  const unsigned int idx4 = blockIdx.x * 256u + tid;  // this thread's float4 idx
  const unsigned int elem = idx4 * 4u;
  if (elem >= n) return;

  // ds_load_b128 from the region this wave just DMA'd.
  const f32x4 xv = *reinterpret_cast<const f32x4*>(&tile[tid * 4u]);

  // fold_in(key(0), 1): all-constant inputs -> constant-folded by clang.
  uint32_t fk0, fk1;
  threefry2x32(0u, 0u, 0u, 1u, fk0, fk1);

  // 512 random bits = 256 draws (4 elements x 64 timesteps x 2 bits).
  uint32_t rb[16];
#pragma unroll
  for (int j = 0; j < 8; ++j)
    threefry2x32(fk0, fk1, idx4, (uint32_t)j, rb[2 * j], rb[2 * j + 1]);

  // First firing timestep: t >= 64*x  <=>  t >= ceil(64*x).
  const int k0 = (int)ceilf(xv.x * 64.0f);
  const int k1 = (int)ceilf(xv.y * 64.0f);
  const int k2 = (int)ceilf(xv.z * 64.0f);
  const int k3 = (int)ceilf(xv.w * 64.0f);

  float* optr = out + (size_t)elem;
#pragma unroll 8
  for (int t = 0; t < TIME_STEPS; ++t) {
    const unsigned int sh = (unsigned int)(t & 15) * 2u;
    const unsigned int w  = (unsigned int)(t >> 4);
    f32x4 o;
    o.x = ((t >= k0) && (((rb[w]       >> sh) & 3u) != 0u)) ? 1.0f : 0.0f;
    o.y = ((t >= k1) && (((rb[4u  + w] >> sh) & 3u) != 0u)) ? 1.0f : 0.0f;
    o.z = ((t >= k2) && (((rb[8u  + w] >> sh) & 3u) != 0u)) ? 1.0f : 0.0f;
    o.w = ((t >= k3) && (((rb[12u + w] >> sh) & 3u) != 0u)) ? 1.0f : 0.0f;
    // NT hint: 616MB output stream, never re-read -> bypass cache retention.
    __builtin_nontemporal_store(o, reinterpret_cast<f32x4*>(optr));
    optr += n;  // next timestep plane
  }
}

extern "C" void kernel_launch(void* const* d_in, const int* in_sizes, int n_in,
                              void* d_out, int out_size, void* d_ws,
                              size_t ws_size, hipStream_t stream) {
  (void)n_in; (void)out_size; (void)d_ws; (void)ws_size;
  const float* x  = (const float*)d_in[0];
  float* out      = (float*)d_out;
  const unsigned int n = (unsigned int)in_sizes[0];   // 2,408,448 (divisible by 1024)

  const unsigned int nvec4  = (n + 3u) / 4u;
  const unsigned int blocks = (nvec4 + 255u) / 256u;  // 2352 blocks of 256 threads
  spike_encode_kernel<<<blocks, 256, 0, stream>>>(x, out, n);
}